// CustomLoss_28930899706387
// MI455X (gfx1250) — compile-verified
//
#include <hip/hip_runtime.h>
#include <math.h>

// Problem constants (from reference)
#define BB 8
#define LL 1024
#define DD 256
#define NN 1024          // P
#define TWO_N 2048
#define KP 128           // DD/2 float2 k-pairs
#define INV_T (1.0f / 0.07f)

typedef __attribute__((ext_vector_type(2))) float v2f;
typedef __attribute__((ext_vector_type(8))) float v8f;

// ---------------------------------------------------------------------------
// Kernel 1: per-example row-major nonzero(mask, size=P, fill=0)
// mask[r][c] = (l1[r] == l0[c]);  i=row index, j=col index
// idx layout per example: jarr[1024] then iarr[1024]
// ---------------------------------------------------------------------------
__global__ void build_indices(const int* __restrict__ labelvecs,
                              int* __restrict__ idx) {
    const int b = blockIdx.x;            // example
    const int t = threadIdx.x;           // 0..1023 (row r of mask)
    const int* l0 = labelvecs + (size_t)b * LL;
    const int* l1 = labelvecs + (size_t)(BB + b) * LL;
    int* jarr = idx + (size_t)b * (2 * LL);
    int* iarr = jarr + LL;

    __shared__ int sl0[LL];
    __shared__ int cnts[LL];

    sl0[t] = l0[t];
    __syncthreads();

    const int my = l1[t];
    int cnt = 0;
    for (int c = 0; c < LL; ++c) cnt += (sl0[c] == my) ? 1 : 0;
    cnts[t] = cnt;
    __syncthreads();

    // Hillis-Steele inclusive scan over 1024 counts
    for (int off = 1; off < LL; off <<= 1) {
        int add = (t >= off) ? cnts[t - off] : 0;
        __syncthreads();
        cnts[t] += add;
        __syncthreads();
    }
    const int base = cnts[t] - cnt;      // exclusive prefix for row t

    // fill_value = 0 init
    jarr[t] = 0;
    iarr[t] = 0;
    __syncthreads();

    // ordered scatter (row-major nonzero, truncated at size=P)
    int pos = base;
    for (int c = 0; c < LL; ++c) {
        if (sl0[c] == my) {
            if (pos < NN) { iarr[pos] = t; jarr[pos] = c; }
            ++pos;
        }
    }
}

// ---------------------------------------------------------------------------
// Kernel 2: gather cf rows into pair-interleaved k-major layout:
//   CFP[b][kp][r] = { cf[r][2*kp], cf[r][2*kp+1] }   (float2, r in [0,2048))
//   rows [0,1024)  : e0[j_r]  -> emb[b,0,:,jarr[r]]
//   rows [1024,2N) : e1[i_r]  -> emb[b,1,:,iarr[r-1024]]
// grid.x = BB*KP blocks (one per (b,kp)), 256 threads; coalesced b64 stores.
// ---------------------------------------------------------------------------
__global__ void gather_cfp(const float* __restrict__ emb,
                           const int* __restrict__ idx,
                           v2f* __restrict__ cfp) {
    const int b  = blockIdx.x >> 7;      // example
    const int kp = blockIdx.x & (KP - 1);
    const int d0 = 2 * kp;
    const int* jarr = idx + (size_t)b * (2 * LL);
    const int* iarr = jarr + LL;
    const float* e0a = emb + ((size_t)(b * 2 + 0) * DD + d0) * LL;
    const float* e0b = e0a + LL;
    const float* e1a = emb + ((size_t)(b * 2 + 1) * DD + d0) * LL;
    const float* e1b = e1a + LL;
    v2f* dst = cfp + ((size_t)b * KP + kp) * TWO_N;

    for (int r = threadIdx.x; r < LL; r += 256) {
        const int j = jarr[r];
        v2f v; v.x = e0a[j]; v.y = e0b[j];
        dst[r] = v;
    }
    for (int r = threadIdx.x; r < LL; r += 256) {
        const int i = iarr[r];
        v2f v; v.x = e1a[i]; v.y = e1b[i];
        dst[LL + r] = v;
    }
}

// ---------------------------------------------------------------------------
// Kernel 3: fused Gram + per-lane online softmax statistics.
// 256 threads = 8 waves; each wave owns 16 rows; block covers 128 rows.
// grid.x = BB * (2048/128) = 128 blocks.
// Per 16-col tile: S16x16 via 64x V_WMMA_F32_16X16X4_F32 (2 accumulators),
// then per-lane online max/sum (single v_exp per element, branchless, diag
// excluded from sum but included in max, partner logit captured).
// Cross-lane merge happens ONCE after the column loop.
// rowloss[b*2048 + r] = m/T + log(sumexp) - dot_partner/T
// ---------------------------------------------------------------------------
__global__ void supcon_main(const v2f* __restrict__ cfp,
                            float* __restrict__ rowloss) {
    const int b    = blockIdx.x >> 4;    // example
    const int rb   = blockIdx.x & 15;    // 128-row block within example
    const int wave = threadIdx.x >> 5;
    const int lane = threadIdx.x & 31;
    const int l15  = lane & 15;
    const int hi   = lane >> 4;          // 0: lanes 0-15, 1: lanes 16-31

    const v2f* cfb = cfp + (size_t)b * KP * TWO_N;   // [128][2048] pair-major
    const int rowBase = rb * 128 + wave * 16;
    const int arow = rowBase + l15;                  // A fragment row (M)

    // per-lane online stats for the 8 rows this lane-half owns
    float m[8], sa[8], pd[8];
#pragma unroll
    for (int v = 0; v < 8; ++v) { m[v] = -INFINITY; sa[v] = 0.0f; pd[v] = 0.0f; }

    for (int ct = 0; ct < TWO_N / 16; ++ct) {
        const int colBase = ct * 16;
        const int bcol = colBase + l15;              // B fragment col (N)

        v8f acc0 = {0.f, 0.f, 0.f, 0.f, 0.f, 0.f, 0.f, 0.f};
        v8f acc1 = {0.f, 0.f, 0.f, 0.f, 0.f, 0.f, 0.f, 0.f};
        // K loop: 256 in steps of 8 (two WMMA f32 16x16x4 per iteration)
#pragma unroll 4
        for (int k = 0; k < DD; k += 8) {
            const int kpA = (k >> 1) + hi;           // K = {k+2hi, k+2hi+1}
            const v2f* pA = cfb + (size_t)kpA * TWO_N;
            const v2f* pB = pA + (size_t)2 * TWO_N;  // next k-step (+4 k)
            v2f a0 = pA[arow], b0 = pA[bcol];
            v2f a1 = pB[arow], b1 = pB[bcol];
            acc0 = __builtin_amdgcn_wmma_f32_16x16x4_f32(
                false, a0, false, b0, (short)0, acc0, false, false);
            acc1 = __builtin_amdgcn_wmma_f32_16x16x4_f32(
                false, a1, false, b1, (short)0, acc1, false, false);
        }
        v8f acc = acc0 + acc1;

        // per-lane online update (C/D layout: M = v + 8*hi, N = l15)
#pragma unroll
        for (int v = 0; v < 8; ++v) {
            const float dot = acc[v];
            const int grow = rowBase + v + 8 * hi;
            const int gcol = colBase + l15;
            // d > 0  -> new per-lane max; one exp serves both rescale & term
            const float d = (dot - m[v]) * INV_T;
            const float t = __expf(fminf(d, -d));    // exp(-|d|)
            const bool  up = d > 0.0f;
            float e = up ? 1.0f : t;                 // exp((dot-mnew)/T)
            const float f = up ? t : 1.0f;           // exp((m-mnew)/T)
            e = (grow == gcol) ? 0.0f : e;           // exclude diagonal
            sa[v] = sa[v] * f + e;
            m[v] = fmaxf(m[v], dot);
            pd[v] += (gcol == (grow ^ NN)) ? dot : 0.0f;  // partner logit
        }
    }

    // one-time merge across the 16 lanes of each half, emit per-row loss
#pragma unroll
    for (int v = 0; v < 8; ++v) {
        float M = m[v];
        M = fmaxf(M, __shfl_xor(M, 1, 32));
        M = fmaxf(M, __shfl_xor(M, 2, 32));
        M = fmaxf(M, __shfl_xor(M, 4, 32));
        M = fmaxf(M, __shfl_xor(M, 8, 32));
        float s = sa[v] * __expf((m[v] - M) * INV_T);
        s += __shfl_xor(s, 1, 32);
        s += __shfl_xor(s, 2, 32);
        s += __shfl_xor(s, 4, 32);
        s += __shfl_xor(s, 8, 32);
        float p = pd[v];
        p += __shfl_xor(p, 1, 32);
        p += __shfl_xor(p, 2, 32);
        p += __shfl_xor(p, 4, 32);
        p += __shfl_xor(p, 8, 32);
        if (l15 == 0) {
            const int grow = rowBase + v + 8 * hi;
            rowloss[(size_t)b * TWO_N + grow] =
                M * INV_T + logf(s) - p * INV_T;
        }
    }
}

// ---------------------------------------------------------------------------
// Kernel 4: deterministic fixed-order reduction of 16384 row losses.
// total = sum(rowloss) / 2048   (TEMP/BASE_TEMP == 1)
// ---------------------------------------------------------------------------
__global__ void reduce_loss(const float* __restrict__ rowloss,
                            float* __restrict__ out) {
    __shared__ float sh[256];
    float acc = 0.0f;
    for (int i = threadIdx.x; i < BB * TWO_N; i += 256) acc += rowloss[i];
    sh[threadIdx.x] = acc;
    __syncthreads();
    for (int s = 128; s > 0; s >>= 1) {
        if (threadIdx.x < s) sh[threadIdx.x] += sh[threadIdx.x + s];
        __syncthreads();
    }
    if (threadIdx.x == 0) out[0] = sh[0] * (1.0f / (float)TWO_N);
}

// ---------------------------------------------------------------------------
extern "C" void kernel_launch(void* const* d_in, const int* in_sizes, int n_in,
                              void* d_out, int out_size, void* d_ws, size_t ws_size,
                              hipStream_t stream) {
    (void)in_sizes; (void)n_in; (void)out_size; (void)ws_size;
    const float* emb       = (const float*)d_in[0];   // [8,2,256,1024] f32
    /* d_in[1]: embeddings_mask (bool) — unused by reference */
    const int*   labelvecs = (const int*)d_in[2];     // [16,1024] i32
    float* out = (float*)d_out;

    char* ws = (char*)d_ws;
    v2f*   cfp     = (v2f*)ws;                                      // 16 MB
    int*   idx     = (int*)(ws + (size_t)BB * KP * TWO_N * 8);      // 64 KB
    float* rowloss = (float*)((char*)idx + (size_t)BB * 2 * LL * 4); // 64 KB

    build_indices<<<BB, 1024, 0, stream>>>(labelvecs, idx);
    gather_cfp<<<BB * KP, 256, 0, stream>>>(emb, idx, cfp);
    supcon_main<<<BB * (TWO_N / 128), 256, 0, stream>>>(cfp, rowloss);
    reduce_loss<<<1, 256, 0, stream>>>(rowloss, out);
}